// LSTM_60026462929068
// MI455X (gfx1250) — compile-verified
//
#include <hip/hip_runtime.h>
#include <hip/hip_bf16.h>

// ---------------------------------------------------------------------------
// LSTM forward, T=16384, H=512.
// Phase 1: x_gates[T,4H] = X @ w_ih^T + b_ih  -> fp32 WMMA GEMM (parallel)
// Phase 2: serial scan, w_hh partitioned across 16 WGPs' LDS (fp32 slices),
//          h exchanged through L2 with device-scope atomics + counter barrier.
// ---------------------------------------------------------------------------

#define Hs   512
#define G4   2048          // 4*H
#define Ts   16384
#define NBLK 16            // workgroups in the scan (one per WGP)
#define TPB  256
#define WPAD 514           // padded LDS row stride (floats): 514%64==2 -> conflict-free

typedef __attribute__((ext_vector_type(2))) float v2f;
typedef __attribute__((ext_vector_type(8))) float v8f;

// ---------------------------------------------------------------------------
// Phase 1: fp32 WMMA GEMM.  C[t,n] = sum_k X[t,k] * Wih[n,k] + b[n]
// Both operands are K-major in memory (NT gemm) -> per-lane contiguous v2f
// fragment loads for V_WMMA_F32_16X16X4_F32:
//   A 16x4 : lane m=L&15, pair K = 2*(L>>4) .. +1
//   B 4x16 : lane n=L&15, pair K = 2*(L>>4) .. +1
//   C 16x16: lane n=L&15, VGPR v holds M = v + 8*(L>>4)
// Each wave computes a 32x32 C footprint (2x2 tiles); 8 waves/block cover
// 64(t) x 128(n).  Grid = (2048/128, 16384/64).
// ---------------------------------------------------------------------------
__global__ __launch_bounds__(256)
void lstm_xgates_wmma(const float* __restrict__ X,
                      const float* __restrict__ Wih,
                      const float* __restrict__ bias,
                      float* __restrict__ xg)
{
  const int lane = threadIdx.x & 31;
  const int wave = threadIdx.x >> 5;
  const int wt   = wave >> 2;                 // 0..1 along t
  const int wn   = wave & 3;                  // 0..3 along n
  const int t0   = blockIdx.y * 64 + wt * 32;
  const int n0   = blockIdx.x * 128 + wn * 32;
  const int m    = lane & 15;
  const int kb   = (lane >> 4) * 2;           // K pair base within K=4 step

  v8f acc[2][2];
  #pragma unroll
  for (int ni = 0; ni < 2; ++ni) {
    float b = bias[n0 + ni * 16 + m];
    #pragma unroll
    for (int v = 0; v < 8; ++v) { acc[0][ni][v] = b; acc[1][ni][v] = b; }
  }

  const float* xr0 = X   + (size_t)(t0 + m)      * Hs;
  const float* xr1 = X   + (size_t)(t0 + 16 + m) * Hs;
  const float* wr0 = Wih + (size_t)(n0 + m)      * Hs;
  const float* wr1 = Wih + (size_t)(n0 + 16 + m) * Hs;

  for (int k0 = 0; k0 < Hs; k0 += 4) {
    v2f a0 = *(const v2f*)(xr0 + k0 + kb);
    v2f a1 = *(const v2f*)(xr1 + k0 + kb);
    v2f b0 = *(const v2f*)(wr0 + k0 + kb);
    v2f b1 = *(const v2f*)(wr1 + k0 + kb);
    acc[0][0] = __builtin_amdgcn_wmma_f32_16x16x4_f32(false, a0, false, b0,
                    (short)0, acc[0][0], false, false);
    acc[0][1] = __builtin_amdgcn_wmma_f32_16x16x4_f32(false, a0, false, b1,
                    (short)0, acc[0][1], false, false);
    acc[1][0] = __builtin_amdgcn_wmma_f32_16x16x4_f32(false, a1, false, b0,
                    (short)0, acc[1][0], false, false);
    acc[1][1] = __builtin_amdgcn_wmma_f32_16x16x4_f32(false, a1, false, b1,
                    (short)0, acc[1][1], false, false);
  }

  const int mhi = (lane >> 4) * 8;
  #pragma unroll
  for (int ti = 0; ti < 2; ++ti)
    #pragma unroll
    for (int ni = 0; ni < 2; ++ni)
      #pragma unroll
      for (int v = 0; v < 8; ++v) {
        int row = t0 + ti * 16 + mhi + v;
        int col = n0 + ni * 16 + m;
        xg[(size_t)row * G4 + col] = acc[ti][ni][v];
      }
}

// ---------------------------------------------------------------------------
// Phase 2: serial scan. Block b owns h[32b..32b+32) and the 128 gate rows
// {q*512 + 32b + j : q=0..3 (i,f,g,o), j=0..31}. Its 128x512 fp32 w_hh slice
// lives in LDS (padded stride 514 -> conflict-free streaming).
// Per step: 256 threads compute 128 dots (two k-halves each), gates, then
// 32 threads update c,h; h-slice stored device-scope into out[t]; all blocks
// sync on a monotonic arrival counter (reset per launch via memsetAsync).
// ---------------------------------------------------------------------------
__device__ __forceinline__ float sigf(float x) {
  return 1.0f / (1.0f + __expf(-x));
}

__global__ __launch_bounds__(TPB)
void lstm_scan(const float* __restrict__ xg,
               const float* __restrict__ Whh,
               float* __restrict__ out,
               unsigned int* __restrict__ cnt)
{
  __shared__ float wl[128 * WPAD];   // 263168 B
  __shared__ float hl[Hs];           //   2048 B
  __shared__ float pl[TPB];          //   1024 B
  __shared__ float gl[128];          //    512 B   (total ~266.8 KB < 320 KB)

  const int tid = threadIdx.x;
  const int blk = blockIdx.x;

  // One-time: stage this block's w_hh slice into LDS.
  for (int idx = tid; idx < 128 * Hs; idx += TPB) {
    int r = idx >> 9;                      // local row 0..127
    int k = idx & (Hs - 1);
    int grow = (r >> 5) * Hs + blk * 32 + (r & 31);   // global gate row
    wl[r * WPAD + k] = Whh[(size_t)grow * Hs + k];
  }

  const int r  = tid & 127;                 // my gate row (local)
  const int kh = (tid >> 7) << 8;           // my k half: 0 or 256
  const int myrow = (r >> 5) * Hs + blk * 32 + (r & 31);

  float c = 0.0f;                           // cell state (tid < 32 only)
  unsigned int target = 0;

  __syncthreads();

  for (int t = 0; t < Ts; ++t) {
    // Gather h_{t-1} (device-scope loads bypass WGP$; out[] is the h history).
    for (int k = tid; k < Hs; k += TPB)
      hl[k] = (t == 0) ? 0.0f
            : __hip_atomic_load(out + (size_t)(t - 1) * Hs + k,
                                __ATOMIC_RELAXED, __HIP_MEMORY_SCOPE_AGENT);
    if (t + 1 < Ts && tid < 128)
      __builtin_prefetch(xg + (size_t)(t + 1) * G4 + myrow, 0, 1);
    __syncthreads();

    // 128 dot products, split into two k-halves across 256 threads.
    {
      const float* wr = wl + r * WPAD + kh;
      const float* hr = hl + kh;
      float a0 = 0.f, a1 = 0.f, a2 = 0.f, a3 = 0.f;
      #pragma unroll 8
      for (int k = 0; k < 256; k += 4) {
        a0 += hr[k]     * wr[k];
        a1 += hr[k + 1] * wr[k + 1];
        a2 += hr[k + 2] * wr[k + 2];
        a3 += hr[k + 3] * wr[k + 3];
      }
      pl[tid] = (a0 + a1) + (a2 + a3);
    }
    __syncthreads();

    if (tid < 128)
      gl[tid] = pl[tid] + pl[tid + 128] + xg[(size_t)t * G4 + myrow];
    __syncthreads();

    if (tid < 32) {
      float ig = sigf(gl[tid]);
      float fg = sigf(gl[32 + tid]);
      float gg = tanhf(gl[64 + tid]);
      float og = sigf(gl[96 + tid]);
      c = fg * c + ig * gg;
      float h = og * tanhf(c);
      __hip_atomic_store(out + (size_t)t * Hs + blk * 32 + tid, h,
                         __ATOMIC_RELAXED, __HIP_MEMORY_SCOPE_AGENT);
    }
    __threadfence();          // own stores visible device-wide
    __syncthreads();

    // Cross-WGP step barrier: monotonic arrival counter through L2.
    target += NBLK;
    if (tid == 0) {
      __hip_atomic_fetch_add(cnt, 1u, __ATOMIC_RELEASE,
                             __HIP_MEMORY_SCOPE_AGENT);
      while (__hip_atomic_load(cnt, __ATOMIC_ACQUIRE,
                               __HIP_MEMORY_SCOPE_AGENT) < target)
        __builtin_amdgcn_s_sleep(2);
    }
    // Hardware cluster barrier (NOP when not dispatched as a cluster);
    // this is the native CDNA5 path for this sync when cluster launch lands.
    __builtin_amdgcn_s_cluster_barrier();
    __syncthreads();
  }
}

extern "C" void kernel_launch(void* const* d_in, const int* in_sizes, int n_in,
                              void* d_out, int out_size, void* d_ws, size_t ws_size,
                              hipStream_t stream) {
  const float* X   = (const float*)d_in[0];   // [T, H]
  const float* Wih = (const float*)d_in[1];   // [4H, H]
  const float* Whh = (const float*)d_in[2];   // [4H, H]
  const float* Bih = (const float*)d_in[3];   // [4H]
  float* out = (float*)d_out;                 // [T, H]

  unsigned int* cnt = (unsigned int*)d_ws;                 // 4 B counter
  float* xg = (float*)((char*)d_ws + 256);                 // [T, 4H] scratch

  // Counter must start at 0 every call (launch is deterministic).
  hipMemsetAsync(d_ws, 0, 4, stream);

  dim3 g1(G4 / 128, Ts / 64);
  lstm_xgates_wmma<<<g1, 256, 0, stream>>>(X, Wih, Bih, xg);
  lstm_scan<<<NBLK, TPB, 0, stream>>>(xg, Whh, out, cnt);
}